// GVSLLoss_249108103544
// MI455X (gfx1250) — compile-verified
//
#include <hip/hip_runtime.h>

typedef __attribute__((ext_vector_type(2))) float v2f;
typedef __attribute__((ext_vector_type(8))) float v8f;

#define VOL   2097152   // 128^3
#define PLANE 16384     // 128^2

// ---------------------------------------------------------------- utilities
__device__ __forceinline__ float waveReduceSum(float v) {
#pragma unroll
  for (int o = 16; o > 0; o >>= 1) v += __shfl_down(v, o, 32);
  return v;
}

__global__ void zero_accum_kernel(double* acc) {
  if (threadIdx.x < 8) acc[threadIdx.x] = 0.0;
}

// ------------------------------------------------- pass 1: products + box-W
// One block handles 4 (d,h) lines of 128 along W. Computes I,J,I2,J2,IJ and
// their 9-tap box sums along W via LDS. Output: dst[5][128][128][128].
__global__ void prod_boxw_kernel(const float* __restrict__ I,
                                 const float* __restrict__ J,
                                 float* __restrict__ dst) {
  __shared__ float s[5][128];
  const int w = threadIdx.x;           // 0..127
  const int lo = (w - 4 < 0) ? 0 : w - 4;
  const int hi = (w + 4 > 127) ? 127 : w + 4;
  for (int l = 0; l < 4; ++l) {
    const int line = blockIdx.x * 4 + l;   // d*128 + h
    const int base = line * 128 + w;
    float iv = I[base];
    float jv = J[base];
    s[0][w] = iv;
    s[1][w] = jv;
    s[2][w] = iv * iv;
    s[3][w] = jv * jv;
    s[4][w] = iv * jv;
    __syncthreads();
#pragma unroll
    for (int ch = 0; ch < 5; ++ch) {
      float acc = 0.0f;
      for (int j = lo; j <= hi; ++j) acc += s[ch][j];
      dst[ch * VOL + base] = acc;
    }
    __syncthreads();
  }
}

// ------------------------------------------------- pass 2: box-H via WMMA
// Out[:,w] = Band(128x128) * X[:,w] per (ch,d) plane, Band[h,j]=1 iff |h-j|<=4.
// One wave computes a 16(h) x 64(w) tile: the band A-tiles (6 K-tiles of 4)
// are built once in registers and reused across four 16-wide N tiles ->
// 24 x V_WMMA_F32_16X16X4_F32 per wave. B loads use clamped j (branch-free):
// out-of-band columns have A==0, so clamped/garbage B contributes nothing.
// EXEC is all-1s (exact grid, no early-out).
__global__ void boxh_wmma_kernel(const float* __restrict__ src,
                                 float* __restrict__ dst) {
  const int wave = blockIdx.x * (blockDim.x >> 5) + (threadIdx.x >> 5);
  const int lane = threadIdx.x & 31;
  const int wt = wave & 1;                 // 2 tiles of 64 along W
  const int ht = (wave >> 1) & 7;          // 8 tiles of 16 along H
  const int d  = (wave >> 4) & 127;
  const int ch = wave >> 11;               // 0..4
  const int h0 = ht << 4;
  const int w0 = wt << 6;

  const float* plane = src + ch * VOL + d * PLANE;
  float*       outp  = dst + ch * VOL + d * PLANE;

  const int n  = lane & 15;                // A: M row; B: N column
  const int kb = (lane >> 4) << 1;         // K base for this half (0 / 2)

  // Band A-tiles, computed once per wave.
  v2f a[6];
#pragma unroll
  for (int kk = 0; kk < 6; ++kk) {
#pragma unroll
    for (int e = 0; e < 2; ++e) {
      const int j = (h0 - 4) + (kk << 2) + kb + e;
      const int dh = (h0 + n) - j;
      a[kk][e] = (j >= 0 && j < 128 && dh <= 4 && dh >= -4) ? 1.0f : 0.0f;
    }
  }

  v8f acc[4] = {};
#pragma unroll
  for (int nt = 0; nt < 4; ++nt) {
    const int w = w0 + (nt << 4) + n;
#pragma unroll
    for (int kk = 0; kk < 6; ++kk) {
      v2f b;
#pragma unroll
      for (int e = 0; e < 2; ++e) {
        int j = (h0 - 4) + (kk << 2) + kb + e;
        j = (j < 0) ? 0 : ((j > 127) ? 127 : j);   // clamp: branch-free load
        b[e] = plane[j * 128 + w];
      }
      acc[nt] = __builtin_amdgcn_wmma_f32_16x16x4_f32(
          /*neg_a=*/false, a[kk], /*neg_b=*/false, b,
          /*c_mod=*/(short)0, acc[nt], /*reuse_a=*/false, /*reuse_b=*/false);
    }
  }

  const int hrow = h0 + ((lane >> 4) << 3);   // lanes 16-31 hold rows M+8
#pragma unroll
  for (int nt = 0; nt < 4; ++nt) {
    const int w = w0 + (nt << 4) + n;
#pragma unroll
    for (int v = 0; v < 8; ++v)
      outp[(hrow + v) * 128 + w] = acc[nt][v];
  }
}

// --------------------------------- pass 3: box-D (ring buffer) + NCC reduce
// blockIdx.x = h, blockIdx.y = d-chunk (32 outputs), threadIdx.x = w.
// Sliding window along D: slot (d+4)%9 aliases slot (d-5)%9.
__global__ void boxd_ncc_kernel(const float* __restrict__ src, double* acc) {
  const int h  = blockIdx.x;
  const int d0 = blockIdx.y * 32;
  const int w  = threadIdx.x;
  const int colbase = h * 128 + w;

  float ring[5][9];
  float wsum[5] = {0.f, 0.f, 0.f, 0.f, 0.f};
  const int zmin = (d0 - 4 < 0) ? 0 : d0 - 4;

  for (int z = zmin; z < d0 + 4; ++z) {      // prologue: z in [max(0,d0-4), d0+3]
#pragma unroll
    for (int ch = 0; ch < 5; ++ch) {
      float v = src[ch * VOL + z * PLANE + colbase];
      wsum[ch] += v;
      ring[ch][z % 9] = v;
    }
  }

  const float win = 729.0f;
  const float inv_win = 1.0f / 729.0f;
  float ccsum = 0.0f;

  for (int d = d0; d < d0 + 32; ++d) {
    const int znew = d + 4;
    const int slot = znew % 9;
    const int zold = d - 5;
#pragma unroll
    for (int ch = 0; ch < 5; ++ch) {
      float v = (znew < 128) ? src[ch * VOL + znew * PLANE + colbase] : 0.0f;
      if (zold >= zmin) wsum[ch] -= ring[ch][slot];
      wsum[ch] += v;
      ring[ch][slot] = v;
    }
    const float Is = wsum[0], Js = wsum[1];
    const float I2 = wsum[2], J2 = wsum[3], IJ = wsum[4];
    const float uI = Is * inv_win, uJ = Js * inv_win;
    const float cross = IJ - uJ * Is - uI * Js + uI * uJ * win;
    const float Ivar  = I2 - 2.0f * uI * Is + uI * uI * win;
    const float Jvar  = J2 - 2.0f * uJ * Js + uJ * uJ * win;
    ccsum += cross * cross / (Ivar * Jvar + 1e-5f);
  }

  ccsum = waveReduceSum(ccsum);
  if ((threadIdx.x & 31) == 0) atomicAdd(&acc[0], (double)ccsum);
}

// ---------------------------------------------------------------- MSE
__global__ void mse_kernel(const float* __restrict__ a,
                           const float* __restrict__ b,
                           double* acc, int n) {
  float s = 0.0f;
  for (int i = blockIdx.x * blockDim.x + threadIdx.x; i < n;
       i += gridDim.x * blockDim.x) {
    float d = a[i] - b[i];
    s += d * d;
  }
  s = waveReduceSum(s);
  if ((threadIdx.x & 31) == 0) atomicAdd(&acc[1], (double)s);
}

// ---------------------------------------------------------------- smoothness
__global__ void smooth_kernel(const float* __restrict__ f, double* acc, int n) {
  float sx = 0.0f, sy = 0.0f, sz = 0.0f;
  for (int i = blockIdx.x * blockDim.x + threadIdx.x; i < n;
       i += gridDim.x * blockDim.x) {
    const int w = i & 127;
    const int h = (i >> 7) & 127;
    const int d = (i >> 14) & 127;
    const float f0 = f[i];
    if (w < 127) { float t = f[i + 1]     - f0; sx += t * t; }
    if (h < 127) { float t = f[i + 128]   - f0; sy += t * t; }
    if (d < 127) { float t = f[i + PLANE] - f0; sz += t * t; }
  }
  sx = waveReduceSum(sx);
  sy = waveReduceSum(sy);
  sz = waveReduceSum(sz);
  if ((threadIdx.x & 31) == 0) {
    atomicAdd(&acc[2], (double)sx);
    atomicAdd(&acc[3], (double)sy);
    atomicAdd(&acc[4], (double)sz);
  }
}

// ---------------------------------------------------------------- finalize
__global__ void finalize_kernel(const double* acc, float* out) {
  if (threadIdx.x == 0 && blockIdx.x == 0) {
    const double vol2 = 2.0 * (double)VOL;                 // 4194304
    const double cnt  = 2.0 * 3.0 * 127.0 * 128.0 * 128.0; // per-diff count
    out[0] = (float)(1.0 - acc[0] / vol2);                 // ncc loss
    out[1] = (float)(acc[1] / vol2);                       // mse loss
    out[2] = (float)((acc[2] + acc[3] + acc[4]) / cnt / 3.0);
  }
}

// ---------------------------------------------------------------- launch
extern "C" void kernel_launch(void* const* d_in, const int* in_sizes, int n_in,
                              void* d_out, int out_size, void* d_ws, size_t ws_size,
                              hipStream_t stream) {
  const float* imgsA  = (const float*)d_in[0];   // [2,1,128,128,128]
  const float* reconA = (const float*)d_in[1];   // [2,1,128,128,128]
  const float* warped = (const float*)d_in[2];   // [2,1,128,128,128]
  const float* flow   = (const float*)d_in[3];   // [2,3,128,128,128]
  float* out = (float*)d_out;

  char* ws = (char*)d_ws;
  double* acc = (double*)ws;                       // 5 doubles
  float* bufA = (float*)(ws + 1024);               // 5 * 128^3 f32
  float* bufB = bufA + (size_t)5 * VOL;            // 5 * 128^3 f32

  zero_accum_kernel<<<1, 32, 0, stream>>>(acc);

  for (int n = 0; n < 2; ++n) {
    prod_boxw_kernel<<<128 * 128 / 4, 128, 0, stream>>>(
        warped + (size_t)n * VOL, imgsA + (size_t)n * VOL, bufA);
    // 5 ch * 128 d * 8 htiles * 2 wtiles(64) = 10240 waves, 8 waves/block
    boxh_wmma_kernel<<<1280, 256, 0, stream>>>(bufA, bufB);
    boxd_ncc_kernel<<<dim3(128, 4), 128, 0, stream>>>(bufB, acc);
  }

  mse_kernel<<<2048, 256, 0, stream>>>(imgsA, reconA, acc, 2 * VOL);
  smooth_kernel<<<4096, 256, 0, stream>>>(flow, acc, 6 * VOL);
  finalize_kernel<<<1, 1, 0, stream>>>(acc, out);
}